// ConvertParamsTEtoParams0TEGaussLayer_50002009260130
// MI455X (gfx1250) — compile-verified
//
#include <hip/hip_runtime.h>

typedef __attribute__((ext_vector_type(2))) float v2f;
typedef __attribute__((ext_vector_type(8))) float v8f;

#define TS 17              // padded LDS tile row stride (floats) -> bank-conflict-free columns
#define TILE_F (16 * TS)   // 272 floats per tile
#define NTILES 16
#define WPB 2              // waves (= batches) per block

enum {
  T_PREC = 0, T_CH, T_CHV, T_CHTE, T_CHVTE, T_CAM, T_CV, T_CVTE,
  T_SYM, T_X, T_A0, T_A1, T_A2, T_L, T_B0, T_B1
};

static __device__ inline v8f vzero() {
  v8f z = {0.f, 0.f, 0.f, 0.f, 0.f, 0.f, 0.f, 0.f};
  return z;
}

// A operand chunk c (covers K=4c..4c+3) for V_WMMA_F32_16X16X4_F32.
// ISA layout: lanes 0-15 hold M=lane, VGPR0->K=4c, VGPR1->K=4c+1;
//             lanes 16-31 hold M=lane-16, K=4c+2, 4c+3.
static __device__ inline v2f lds_a(const float* __restrict__ t, int lane, int c, bool transp) {
  int m = lane & 15;
  int k = 4 * c + 2 * (lane >> 4);
  v2f r;
  if (!transp) { r.x = t[m * TS + k];       r.y = t[m * TS + k + 1]; }
  else         { r.x = t[k * TS + m];       r.y = t[(k + 1) * TS + m]; }
  return r;
}

// B operand chunk c: lane = column N, same K split across lane halves.
static __device__ inline v2f lds_b(const float* __restrict__ t, int lane, int c, bool transp) {
  int n = lane & 15;
  int k = 4 * c + 2 * (lane >> 4);
  v2f r;
  if (!transp) { r.x = t[k * TS + n];       r.y = t[(k + 1) * TS + n]; }
  else         { r.x = t[n * TS + k];       r.y = t[n * TS + k + 1]; }
  return r;
}

// C/D layout: VGPR r holds element (M = r + 8*(lane>>4), N = lane&15)
static __device__ inline void store_c(float* __restrict__ t, int lane, v8f v) {
  int n = lane & 15;
  int mo = (lane >> 4) * 8;
  #pragma unroll
  for (int r = 0; r < 8; ++r) t[(mo + r) * TS + n] = v[r];
}

static __device__ inline v8f load_c(const float* __restrict__ t, int lane) {
  int n = lane & 15;
  int mo = (lane >> 4) * 8;
  v8f v;
  #pragma unroll
  for (int r = 0; r < 8; ++r) v[r] = t[(mo + r) * TS + n];
  return v;
}

// acc += op(A) @ op(B) : one 16x16x16 fp32 matmul = 4 x v_wmma_f32_16x16x4_f32
static __device__ inline v8f mm(const float* __restrict__ ta, bool tra,
                                const float* __restrict__ tb, bool trb,
                                int lane, v8f acc) {
  #pragma unroll
  for (int c = 0; c < 4; ++c) {
    v2f a = lds_a(ta, lane, c, tra);
    v2f b = lds_b(tb, lane, c, trb);
    acc = __builtin_amdgcn_wmma_f32_16x16x4_f32(false, a, false, b, (short)0, acc,
                                                false, false);
  }
  return acc;
}

// Stage one 16x16 fp32 matrix global -> LDS tile; 2 coalesced b128 loads/lane.
static __device__ inline void load_tile(float* __restrict__ t, const float* __restrict__ g,
                                        int lane) {
  const float4* g4 = (const float4*)g;
  float4 lo = g4[lane * 2 + 0];
  float4 hi = g4[lane * 2 + 1];
  int row = lane >> 1;
  int c0 = (lane & 1) * 8;
  float* p = t + row * TS + c0;
  p[0] = lo.x; p[1] = lo.y; p[2] = lo.z; p[3] = lo.w;
  p[4] = hi.x; p[5] = hi.y; p[6] = hi.z; p[7] = hi.w;
}

__global__ __launch_bounds__(32 * WPB) void gauss_params_te_kernel(
    const float* __restrict__ mu,          // (B, 32)
    const float* __restrict__ prec_h,      // (B,16,16)
    const float* __restrict__ chol_v,
    const float* __restrict__ chol_h,
    const float* __restrict__ chol_hv,
    const float* __restrict__ chol_v_TE,
    const float* __restrict__ chol_h_TE,
    const float* __restrict__ chol_hv_TE,
    const float* __restrict__ chol_amat,
    float* __restrict__ out, int Btot) {
  __shared__ float smem[WPB * NTILES * TILE_F];
  const int lane = threadIdx.x & 31;
  const int wave = threadIdx.x >> 5;
  const int b = blockIdx.x * WPB + wave;
  if (b >= Btot) return;                    // uniform per wave
  float* tb_ = smem + wave * NTILES * TILE_F;
  #define TL(id) (tb_ + (id) * TILE_F)
  const size_t mb = (size_t)b * 256;

  load_tile(TL(T_PREC),  prec_h     + mb, lane);
  load_tile(TL(T_CH),    chol_h     + mb, lane);
  load_tile(TL(T_CHV),   chol_hv    + mb, lane);
  load_tile(TL(T_CHTE),  chol_h_TE  + mb, lane);
  load_tile(TL(T_CHVTE), chol_hv_TE + mb, lane);
  load_tile(TL(T_CAM),   chol_amat  + mb, lane);
  load_tile(TL(T_CV),    chol_v     + mb, lane);
  load_tile(TL(T_CVTE),  chol_v_TE  + mb, lane);

  const int n = lane & 15;
  const int mo = (lane >> 4) * 8;

  // sym = cHTE@cH^T + cH@cHTE^T + cHVTE@cHV^T + cHV@cHVTE^T
  v8f s = vzero();
  s = mm(TL(T_CHTE),  false, TL(T_CH),    true, lane, s);
  s = mm(TL(T_CH),    false, TL(T_CHTE),  true, lane, s);
  s = mm(TL(T_CHVTE), false, TL(T_CHV),   true, lane, s);
  s = mm(TL(T_CHV),   false, TL(T_CHVTE), true, lane, s);
  store_c(TL(T_SYM), lane, s);

  // Pinv via Newton-Schulz: X <- X(2I - P X); P SPD, spectrum ~[1, 1.7]
  {
    v8f x;
    #pragma unroll
    for (int r = 0; r < 8; ++r) x[r] = ((mo + r) == n) ? 0.65f : 0.0f;
    store_c(TL(T_X), lane, x);
  }
  for (int it = 0; it < 8; ++it) {
    v8f y = mm(TL(T_PREC), false, TL(T_X), false, lane, vzero());   // P@X
    store_c(TL(T_A0), lane, y);
    v8f w = mm(TL(T_X), false, TL(T_A0), false, lane, vzero());     // X@(P@X)
    v8f x = load_c(TL(T_X), lane);
    #pragma unroll
    for (int r = 0; r < 8; ++r) x[r] = 2.0f * x[r] - w[r];
    store_c(TL(T_X), lane, x);
  }

  // prec_h_inv_TE = -(Pinv @ sym @ Pinv) -> overwrite T_SYM
  {
    v8f t = mm(TL(T_X), false, TL(T_SYM), false, lane, vzero());
    store_c(TL(T_A0), lane, t);
    v8f u = mm(TL(T_A0), false, TL(T_X), false, lane, vzero());
    #pragma unroll
    for (int r = 0; r < 8; ++r) u[r] = -u[r];
    store_c(TL(T_SYM), lane, u);
  }

  // t1 = Pinv@chol_hv, t2 = pite@chol_hv, t3 = Pinv@chol_hv_TE
  store_c(TL(T_A0), lane, mm(TL(T_X),   false, TL(T_CHV),   false, lane, vzero()));
  store_c(TL(T_A1), lane, mm(TL(T_SYM), false, TL(T_CHV),   false, lane, vzero()));
  store_c(TL(T_A2), lane, mm(TL(T_X),   false, TL(T_CHVTE), false, lane, vzero()));

  // amat_TE = -(cHVTE^T@t1 + cHV^T@t2 + cHV^T@t3) -> T_A0
  {
    v8f a = vzero();
    a = mm(TL(T_CHVTE), true, TL(T_A0), false, lane, a);
    a = mm(TL(T_CHV),   true, TL(T_A1), false, lane, a);
    a = mm(TL(T_CHV),   true, TL(T_A2), false, lane, a);
    #pragma unroll
    for (int r = 0; r < 8; ++r) a[r] = -a[r];
    store_c(TL(T_A0), lane, a);
  }

  // cinv = inv(I + L), L strictly lower (nilpotent): (I-L)(I+L^2)(I+L^4)(I+L^8)
  v8f lv = load_c(TL(T_CAM), lane);
  #pragma unroll
  for (int r = 0; r < 8; ++r) if ((mo + r) == n) lv[r] = 0.0f;
  store_c(TL(T_L), lane, lv);
  store_c(TL(T_B0), lane, mm(TL(T_L),  false, TL(T_L),  false, lane, vzero()));  // L^2
  store_c(TL(T_B1), lane, mm(TL(T_B0), false, TL(T_B0), false, lane, vzero()));  // L^4
  store_c(TL(T_A1), lane, mm(TL(T_B1), false, TL(T_B1), false, lane, vzero()));  // L^8
  {
    v8f im;
    #pragma unroll
    for (int r = 0; r < 8; ++r) im[r] = (((mo + r) == n) ? 1.0f : 0.0f) - lv[r];
    store_c(TL(T_A2), lane, im);                                    // I - L
    v8f m1 = mm(TL(T_A2), false, TL(T_B0), false, lane, im);        // (I-L)(I+L^2)
    store_c(TL(T_A2), lane, m1);
    v8f m2 = mm(TL(T_A2), false, TL(T_B1), false, lane, m1);        // *(I+L^4)
    store_c(TL(T_A2), lane, m2);
    v8f ci = mm(TL(T_A2), false, TL(T_A1), false, lane, m2);        // *(I+L^8)
    store_c(TL(T_A2), lane, ci);                                    // cinv
  }

  // arg = (cinv @ amat_TE @ cinv^T) * phi_mask -> T_B0
  store_c(TL(T_L), lane, mm(TL(T_A2), false, TL(T_A0), false, lane, vzero()));
  {
    v8f arg = mm(TL(T_L), false, TL(T_A2), true, lane, vzero());
    #pragma unroll
    for (int r = 0; r < 8; ++r) {
      int m = mo + r;
      float msk = (m > n) ? 1.0f : ((m == n) ? 0.5f : 0.0f);
      arg[r] *= msk;
    }
    store_c(TL(T_B0), lane, arg);
  }
  // chol_amat_TE = chol_amat @ masked_arg -> T_B1
  store_c(TL(T_B1), lane, mm(TL(T_CAM), false, TL(T_B0), false, lane, vzero()));

  // cholv_TE_std = chol_v_TE @ chol_amat + chol_v @ chol_amat_TE
  v8f res = mm(TL(T_CVTE), false, TL(T_CAM), false, lane, vzero());
  res = mm(TL(T_CV), false, TL(T_B1), false, lane, res);

  float* o = out + mb;
  #pragma unroll
  for (int r = 0; r < 8; ++r) o[(mo + r) * 16 + n] = res[r];

  // muv_TE = mu_TE[:, :16]  (second output, concatenated after first)
  if (lane < 16) out[(size_t)Btot * 256 + (size_t)b * 16 + lane] = mu[(size_t)b * 32 + lane];
  #undef TL
}

extern "C" void kernel_launch(void* const* d_in, const int* in_sizes, int n_in,
                              void* d_out, int out_size, void* d_ws, size_t ws_size,
                              hipStream_t stream) {
  const float* mu         = (const float*)d_in[0];
  const float* prec_h     = (const float*)d_in[1];
  const float* chol_v     = (const float*)d_in[2];
  const float* chol_h     = (const float*)d_in[3];
  const float* chol_hv    = (const float*)d_in[4];
  const float* chol_v_TE  = (const float*)d_in[5];
  const float* chol_h_TE  = (const float*)d_in[6];
  const float* chol_hv_TE = (const float*)d_in[7];
  const float* chol_amat  = (const float*)d_in[8];
  float* out = (float*)d_out;
  (void)n_in; (void)out_size; (void)d_ws; (void)ws_size;

  int Btot = in_sizes[0] / 32;   // mu_TE is (B, 32)
  dim3 grid((Btot + WPB - 1) / WPB), block(32 * WPB);
  gauss_params_te_kernel<<<grid, block, 0, stream>>>(
      mu, prec_h, chol_v, chol_h, chol_hv, chol_v_TE, chol_h_TE, chol_hv_TE,
      chol_amat, out, Btot);
}